// Model_12249246728722
// MI455X (gfx1250) — compile-verified
//
#include <hip/hip_runtime.h>
#include <hip/hip_bf16.h>

// Problem constants (from reference)
#define Bsz  16
#define Tt   512
#define Cc   1024
#define Dd   4096
#define Kk   128
#define NW   127          // num_windows = (512-8)/4 + 1
#define NTOK 8192         // B*T

typedef __attribute__((ext_vector_type(2))) float v2f;
typedef __attribute__((ext_vector_type(8))) float v8f;

// ---------------------------------------------------------------------------
// helpers
// ---------------------------------------------------------------------------
__device__ __forceinline__ unsigned f2key(float f) {
    unsigned u = __float_as_uint(f);
    return u ^ ((u >> 31) ? 0xFFFFFFFFu : 0x80000000u);   // monotone: bigger float -> bigger key
}

// inclusive block scan over 256 threads; caller may read sc[255] (=total)
// immediately after return; function self-syncs at entry so sc can be reused.
__device__ __forceinline__ int block_scan_incl(int* sc, int tid, int v) {
    __syncthreads();
    sc[tid] = v; __syncthreads();
#pragma unroll
    for (int o = 1; o < 256; o <<= 1) {
        int x = sc[tid];
        int y = (tid >= o) ? sc[tid - o] : 0;
        __syncthreads();
        sc[tid] = x + y;
        __syncthreads();
    }
    return sc[tid];
}

// radix-select (descending) threshold of the k-th largest among n u32 keys in LDS.
// On return: *thrOut = exact key value of the k-th largest; *needOut = number of
// elements equal to thr that belong to the top-k (ties to take, index-ordered).
__device__ __forceinline__ void radix_thr(const unsigned* keys, unsigned* hist,
                                          int* sBin, int* sNeed,
                                          int n, int k, int tid,
                                          unsigned* thrOut, int* needOut) {
    unsigned prefix = 0, pmask = 0;
    int need = k;
    for (int shift = 24; shift >= 0; shift -= 8) {
        hist[tid] = 0;
        __syncthreads();
        for (int i = tid; i < n; i += 256) {
            unsigned kv = keys[i];
            if ((kv & pmask) == prefix) atomicAdd(&hist[(kv >> shift) & 0xFFu], 1u);
        }
        __syncthreads();
        if (tid == 0) {
            int cum = 0, bin = 0, nd = need;
            for (int bv = 255; bv >= 0; --bv) {
                int c = (int)hist[bv];
                if (cum + c >= need) { bin = bv; nd = need - cum; break; }
                cum += c;
            }
            *sBin = bin; *sNeed = nd;
        }
        __syncthreads();
        prefix |= ((unsigned)(*sBin)) << shift;
        pmask  |= 0xFFu << shift;
        need    = *sNeed;
        __syncthreads();
    }
    *thrOut = prefix; *needOut = need;
}

// ---------------------------------------------------------------------------
// K0: zero accumulators (pooled + loss) each launch
// ---------------------------------------------------------------------------
__global__ void k_zero(float* __restrict__ pooled, int n, float* __restrict__ lossAcc) {
    int i = blockIdx.x * blockDim.x + threadIdx.x;
    if (i < n) pooled[i] = 0.f;
    if (i == 0) *lossAcc = 0.f;
}

// ---------------------------------------------------------------------------
// K1: encoder GEMM  post = relu((x - b_dec) @ W_enc^T + b_enc)
//     f32 WMMA 16x16x4. Block tile 128x128 (4 waves), wave tile 64x64
//     (16 v8f accumulators) -> 16 WMMAs per 8 LDS fragment loads.
//     grid = (Dd/128, NTOK/128), block = 128 threads
// ---------------------------------------------------------------------------
#define KC 32
#define LDST 36   // LDS row stride (floats): conflict-free frags; 144B pitch keeps b128 stores aligned

__global__ __launch_bounds__(128) void
k_encode(const float* __restrict__ x, const float* __restrict__ Wenc,
         const float* __restrict__ b_dec, const float* __restrict__ b_enc,
         float* __restrict__ post) {
    __shared__ float At[128 * LDST];
    __shared__ float Bt[128 * LDST];

    const int tid  = threadIdx.x;
    const int lane = tid & 31;
    const int wv   = tid >> 5;          // wave id 0..3
    const int lm   = lane & 15;
    const int lh   = lane >> 4;
    const int wm   = (wv & 1) * 64;     // wave M offset within block tile
    const int wn   = (wv >> 1) * 64;    // wave N offset within block tile

    const int nBase = blockIdx.x * 128; // D dimension
    const int mBase = blockIdx.y * 128; // token dimension

    v8f acc[4][4];
#pragma unroll
    for (int ms = 0; ms < 4; ++ms)
#pragma unroll
        for (int ns = 0; ns < 4; ++ns) acc[ms][ns] = v8f{};

    const int r = tid;                  // staged row 0..127 (one row per thread)

    for (int c0 = 0; c0 < Cc; c0 += KC) {
        // ---- stage A (x - b_dec) and B (W_enc) tiles: 128 rows x 32 cols each
#pragma unroll
        for (int i = 0; i < 8; ++i) {
            const int col = i * 4;
            float4 a  = *(const float4*)(x     + (size_t)(mBase + r) * Cc + c0 + col);
            float4 bd = *(const float4*)(b_dec + c0 + col);
            float4 w  = *(const float4*)(Wenc  + (size_t)(nBase + r) * Cc + c0 + col);
            At[r * LDST + col + 0] = a.x - bd.x;
            At[r * LDST + col + 1] = a.y - bd.y;
            At[r * LDST + col + 2] = a.z - bd.z;
            At[r * LDST + col + 3] = a.w - bd.w;
            Bt[r * LDST + col + 0] = w.x;
            Bt[r * LDST + col + 1] = w.y;
            Bt[r * LDST + col + 2] = w.z;
            Bt[r * LDST + col + 3] = w.w;
        }
        __syncthreads();

        // ---- compute: 8 K-steps x (4 A-frags + 4 B-frags -> 16 WMMAs)
#pragma unroll
        for (int kk = 0; kk < KC; kk += 4) {
            v2f af[4], bf[4];
#pragma unroll
            for (int ms = 0; ms < 4; ++ms)
                af[ms] = *(const v2f*)&At[(wm + 16 * ms + lm) * LDST + kk + 2 * lh];
#pragma unroll
            for (int ns = 0; ns < 4; ++ns)
                bf[ns] = *(const v2f*)&Bt[(wn + 16 * ns + lm) * LDST + kk + 2 * lh];
#pragma unroll
            for (int ms = 0; ms < 4; ++ms)
#pragma unroll
                for (int ns = 0; ns < 4; ++ns)
                    acc[ms][ns] = __builtin_amdgcn_wmma_f32_16x16x4_f32(
                        false, af[ms], false, bf[ns], (short)0, acc[ms][ns], false, false);
        }
        __syncthreads();
    }

    // ---- epilogue: D layout -> VGPR g holds row M = g + 8*lh, col N = lm
#pragma unroll
    for (int ns = 0; ns < 4; ++ns) {
        const int col = nBase + wn + 16 * ns + lm;
        const float bias = b_enc[col];
#pragma unroll
        for (int ms = 0; ms < 4; ++ms) {
#pragma unroll
            for (int g = 0; g < 8; ++g) {
                const int row = mBase + wm + 16 * ms + g + 8 * lh;
                float v = acc[ms][ns][g] + bias;
                post[(size_t)row * Dd + col] = v > 0.f ? v : 0.f;
            }
        }
    }
}

// ---------------------------------------------------------------------------
// K2: window sums + per-window top-128 -> bitmask. grid = B*NW, block = 256
// ---------------------------------------------------------------------------
__global__ __launch_bounds__(256) void
k_window_mask(const float* __restrict__ post, unsigned* __restrict__ maskw) {
    __shared__ unsigned keys[Dd];
    __shared__ unsigned hist[256];
    __shared__ int sc[256];
    __shared__ unsigned mword[Dd / 32];
    __shared__ int sBin, sNeed, sTieBase;

    const int tid = threadIdx.x;
    const int blk = blockIdx.x;
    const int b = blk / NW, w = blk % NW;
    const float* prow = post + (size_t)(b * Tt + w * 4) * Dd;

    for (int d = tid; d < Dd; d += 256) {
        float s = 0.f;
#pragma unroll
        for (int i = 0; i < 8; ++i) s += prow[(size_t)i * Dd + d];
        keys[d] = f2key(s);
    }
    for (int i = tid; i < Dd / 32; i += 256) mword[i] = 0u;
    __syncthreads();

    unsigned thr; int need;
    radix_thr(keys, hist, &sBin, &sNeed, Dd, Kk, tid, &thr, &need);

    if (tid == 0) sTieBase = 0;
    __syncthreads();
    for (int base = 0; base < Dd; base += 256) {
        unsigned kv = keys[base + tid];
        int isTie = (kv == thr) ? 1 : 0;
        int incl  = block_scan_incl(sc, tid, isTie);
        int tot   = sc[255];
        int rank  = sTieBase + incl - isTie;
        if ((kv > thr) || (isTie && rank < need))
            atomicOr(&mword[(base + tid) >> 5], 1u << (tid & 31));
        __syncthreads();
        if (tid == 0) sTieBase += tot;
        __syncthreads();
    }
    for (int i = tid; i < Dd / 32; i += 256)
        maskw[(size_t)blk * (Dd / 32) + i] = mword[i];
}

// ---------------------------------------------------------------------------
// K3: feature votes = post * cover-count; per-token top-128 -> sparse (idx,val)
//     + fused pooled mean accumulation. grid = NTOK, block = 256
// ---------------------------------------------------------------------------
__global__ __launch_bounds__(256) void
k_select(const float* __restrict__ post, const unsigned* __restrict__ maskw,
         int* __restrict__ idxOut, float* __restrict__ valOut,
         float* __restrict__ pooled) {
    __shared__ unsigned keys[Dd];
    __shared__ unsigned hist[256];
    __shared__ int sc[256];
    __shared__ unsigned wm[2][Dd / 32];
    __shared__ int sBin, sNeed, sTieBase, sSelBase;

    const int tid = threadIdx.x;
    const int n = blockIdx.x;
    const int b = n / Tt, t = n % Tt;
    int wlo = (t >= 4) ? ((t - 4) >> 2) : 0;   // = max(0, ceil((t-7)/4))
    int whi = t >> 2; if (whi > NW - 1) whi = NW - 1;
    const int nw = whi - wlo + 1;              // 1 or 2 covering windows

    for (int i = tid; i < nw * (Dd / 32); i += 256)
        wm[i / (Dd / 32)][i % (Dd / 32)] =
            maskw[(size_t)(b * NW + wlo + i / (Dd / 32)) * (Dd / 32) + (i % (Dd / 32))];
    __syncthreads();

    const float* prow = post + (size_t)n * Dd;
    for (int d = tid; d < Dd; d += 256) {
        unsigned word = d >> 5, bit = 1u << (d & 31);
        int cover = ((wm[0][word] & bit) ? 1 : 0) + ((nw > 1 && (wm[1][word] & bit)) ? 1 : 0);
        keys[d] = f2key(prow[d] * (float)cover);
    }
    __syncthreads();

    unsigned thr; int need;
    radix_thr(keys, hist, &sBin, &sNeed, Dd, Kk, tid, &thr, &need);

    if (tid == 0) { sTieBase = 0; sSelBase = 0; }
    __syncthreads();
    for (int base = 0; base < Dd; base += 256) {
        unsigned kv = keys[base + tid];
        int isTie = (kv == thr) ? 1 : 0;
        int inclT = block_scan_incl(sc, tid, isTie);
        int totT  = sc[255];
        int rank  = sTieBase + inclT - isTie;
        int sel   = ((kv > thr) || (isTie && rank < need)) ? 1 : 0;
        int inclS = block_scan_incl(sc, tid, sel);
        int totS  = sc[255];
        int slot  = sSelBase + inclS - sel;
        if (sel) {
            int d = base + tid;
            float v = prow[d];
            idxOut[(size_t)n * Kk + slot] = d;
            valOut[(size_t)n * Kk + slot] = v;
            atomicAdd(&pooled[(size_t)b * Dd + d], v * (1.0f / (float)Tt));
        }
        __syncthreads();
        if (tid == 0) { sTieBase += totT; sSelBase += totS; }
        __syncthreads();
    }
}

// ---------------------------------------------------------------------------
// K4: sparse decode (128 W_enc-row AXPYs, L2-resident) + fused MSE loss.
//     grid = NTOK, block = 256 (each thread owns 4 of the 1024 channels)
// ---------------------------------------------------------------------------
__global__ __launch_bounds__(256) void
k_decode_loss(const float* __restrict__ x, const float* __restrict__ Wenc,
              const float* __restrict__ b_dec, const int* __restrict__ idx,
              const float* __restrict__ val, float* __restrict__ lossAcc) {
    __shared__ int   sIdx[Kk];
    __shared__ float sVal[Kk];
    __shared__ float red[256];

    const int tid = threadIdx.x;
    const int n = blockIdx.x;
    if (tid < Kk) {
        sIdx[tid] = idx[(size_t)n * Kk + tid];
        sVal[tid] = val[(size_t)n * Kk + tid];
    }
    __syncthreads();

    float acc[4];
#pragma unroll
    for (int j = 0; j < 4; ++j) acc[j] = b_dec[tid + 256 * j];

    for (int i = 0; i < Kk; ++i) {
        const float* wrow = Wenc + (size_t)sIdx[i] * Cc;   // W_dec[c][d] == W_enc[d][c]
        const float  v = sVal[i];
#pragma unroll
        for (int j = 0; j < 4; ++j) acc[j] += v * wrow[tid + 256 * j];
    }

    const float* xrow = x + (size_t)n * Cc;
    float e = 0.f;
#pragma unroll
    for (int j = 0; j < 4; ++j) { float d = acc[j] - xrow[tid + 256 * j]; e += d * d; }

    red[tid] = e; __syncthreads();
    for (int o = 128; o > 0; o >>= 1) { if (tid < o) red[tid] += red[tid + o]; __syncthreads(); }
    if (tid == 0) atomicAdd(lossAcc, red[0]);
}

// ---------------------------------------------------------------------------
// K5: LayerNorm + MLP head + log_softmax + final loss write. grid = B, block 256
// ---------------------------------------------------------------------------
__global__ __launch_bounds__(256) void
k_head(const float* __restrict__ pooled, const float* __restrict__ ln_g,
       const float* __restrict__ ln_b, const float* __restrict__ W1,
       const float* __restrict__ b1, const float* __restrict__ W2,
       const float* __restrict__ b2, const float* __restrict__ lossAcc,
       float* __restrict__ out) {
    __shared__ float lnv[Dd];
    __shared__ float red[256];
    __shared__ float hs[256];
    __shared__ float sMu, sRstd, sL0, sL1;

    const int tid = threadIdx.x;
    const int b = blockIdx.x;
    const float* prow = pooled + (size_t)b * Dd;

    float s = 0.f, ss = 0.f;
    for (int d = tid; d < Dd; d += 256) { float p = prow[d]; s += p; ss += p * p; }
    red[tid] = s; __syncthreads();
    for (int o = 128; o > 0; o >>= 1) { if (tid < o) red[tid] += red[tid + o]; __syncthreads(); }
    if (tid == 0) sMu = red[0] / (float)Dd;
    __syncthreads();
    red[tid] = ss; __syncthreads();
    for (int o = 128; o > 0; o >>= 1) { if (tid < o) red[tid] += red[tid + o]; __syncthreads(); }
    if (tid == 0) {
        float mu = sMu;
        float var = red[0] / (float)Dd - mu * mu;
        sRstd = rsqrtf(var + 1e-5f);
    }
    __syncthreads();
    const float mu = sMu, rstd = sRstd;
    for (int d = tid; d < Dd; d += 256)
        lnv[d] = (prow[d] - mu) * rstd * ln_g[d] + ln_b[d];
    __syncthreads();

    float acc = b1[tid];
    const float* wrow = W1 + (size_t)tid * Dd;
    for (int d = 0; d < Dd; ++d) acc += lnv[d] * wrow[d];   // lnv broadcast from LDS
    hs[tid] = acc > 0.f ? acc : 0.f;
    __syncthreads();

    red[tid] = hs[tid] * W2[tid]; __syncthreads();
    for (int o = 128; o > 0; o >>= 1) { if (tid < o) red[tid] += red[tid + o]; __syncthreads(); }
    if (tid == 0) sL0 = red[0];
    __syncthreads();
    red[tid] = hs[tid] * W2[256 + tid]; __syncthreads();
    for (int o = 128; o > 0; o >>= 1) { if (tid < o) red[tid] += red[tid + o]; __syncthreads(); }
    if (tid == 0) sL1 = red[0];
    __syncthreads();

    if (tid == 0) {
        float l0 = sL0 + b2[0], l1 = sL1 + b2[1];
        float m = fmaxf(l0, l1);
        float lse = m + logf(expf(l0 - m) + expf(l1 - m));
        out[2 * b + 0] = l0 - lse;
        out[2 * b + 1] = l1 - lse;
        if (b == 0) out[32] = *lossAcc * (1.0f / ((float)NTOK * (float)Cc));
    }
}

// ---------------------------------------------------------------------------
// launch
// ---------------------------------------------------------------------------
extern "C" void kernel_launch(void* const* d_in, const int* in_sizes, int n_in,
                              void* d_out, int out_size, void* d_ws, size_t ws_size,
                              hipStream_t stream) {
    (void)in_sizes; (void)n_in; (void)out_size; (void)ws_size;
    const float* x    = (const float*)d_in[0];
    const float* Wenc = (const float*)d_in[1];
    const float* benc = (const float*)d_in[2];
    const float* bdec = (const float*)d_in[4];
    const float* lng  = (const float*)d_in[5];
    const float* lnb  = (const float*)d_in[6];
    const float* W1   = (const float*)d_in[7];
    const float* b1   = (const float*)d_in[8];
    const float* W2   = (const float*)d_in[9];
    const float* b2   = (const float*)d_in[10];
    float* out = (float*)d_out;

    char* ws = (char*)d_ws;
    size_t off = 0;
    auto carve = [&](size_t bytes) {
        void* p = ws + off;
        off = (off + bytes + 255) & ~(size_t)255;
        return p;
    };
    float*    post    = (float*)   carve((size_t)NTOK * Dd * sizeof(float));     // 128 MiB
    unsigned* maskw   = (unsigned*)carve((size_t)Bsz * NW * (Dd / 32) * 4);      // ~1 MiB
    int*      idx     = (int*)     carve((size_t)NTOK * Kk * sizeof(int));       // 4 MiB
    float*    val     = (float*)   carve((size_t)NTOK * Kk * sizeof(float));     // 4 MiB
    float*    pooled  = (float*)   carve((size_t)Bsz * Dd * sizeof(float));      // 256 KiB
    float*    lossAcc = (float*)   carve(256);

    k_zero<<<dim3((Bsz * Dd + 255) / 256), 256, 0, stream>>>(pooled, Bsz * Dd, lossAcc);
    k_encode<<<dim3(Dd / 128, NTOK / 128), 128, 0, stream>>>(x, Wenc, bdec, benc, post);
    k_window_mask<<<dim3(Bsz * NW), 256, 0, stream>>>(post, maskw);
    k_select<<<dim3(NTOK), 256, 0, stream>>>(post, maskw, idx, val, pooled);
    k_decode_loss<<<dim3(NTOK), 256, 0, stream>>>(x, Wenc, bdec, idx, val, lossAcc);
    k_head<<<dim3(Bsz), 256, 0, stream>>>(pooled, lng, lnb, W1, b1, W2, b2, lossAcc, out);
}